// WBM_86251533238490
// MI455X (gfx1250) — compile-verified
//
#include <hip/hip_runtime.h>
#include <stdint.h>

typedef float v2f __attribute__((ext_vector_type(2)));
typedef float v8f __attribute__((ext_vector_type(8)));

#define HW 262144
#define IMG_W 512

#define AS1 __attribute__((address_space(1)))
#define AS3 __attribute__((address_space(3)))

#if defined(__has_builtin)
#if __has_builtin(__builtin_amdgcn_global_load_async_to_lds_b32) && \
    __has_builtin(__builtin_amdgcn_s_wait_asynccnt)
#define HAVE_ASYNC_LDS 1
#endif
#endif

// ---------- workspace layout ----------
constexpr size_t OFF_H1   = 0;                          // 192*2048 u32
constexpr size_t OFF_H2   = OFF_H1 + 192ull*2048*4;     // 192*4*2048 u32
constexpr size_t OFF_H3   = OFF_H2 + 192ull*4*2048*4;   // 192*4*1024 u32
constexpr size_t OFF_ZEND = OFF_H3 + 192ull*4*1024*4;   // zero-init up to here
constexpr size_t OFF_SELP = OFF_ZEND;                   // 192*4 u32
constexpr size_t OFF_SELR = OFF_SELP + 192ull*4*4;      // 192*4 u32
constexpr size_t OFF_WBM  = OFF_SELR + 192ull*4*4;      // 192 f32
constexpr size_t OFF_WBS  = OFF_WBM + 192ull*4;         // 192 f32
constexpr size_t OFF_MEAN = OFF_WBS + 192ull*4;         // 64 f32
constexpr size_t OFF_RS   = OFF_MEAN + 64ull*4;         // 64 f32
constexpr size_t OFF_P1   = OFF_RS + 64ull*4;           // 32768 f32
constexpr size_t OFF_P2   = OFF_P1 + 32768ull*4;        // 32768 f32

__device__ __forceinline__ uint32_t fkey(float f) {
  uint32_t u = __float_as_uint(f);
  uint32_t mask = (uint32_t)((int32_t)u >> 31) | 0x80000000u;
  return u ^ mask;
}
__device__ __forceinline__ float keyval(uint32_t k) {
  uint32_t mask = (k & 0x80000000u) ? 0x80000000u : 0xFFFFFFFFu;
  return __uint_as_float(k ^ mask);
}
__device__ __forceinline__ const float* pick(const float* a, const float* b,
                                             const float* c, const float* d, int i) {
  return i == 0 ? a : (i == 1 ? b : (i == 2 ? c : d));
}

// Compile-time tap -> LDS offset (relative to tile row 0, col 0 of the block).
// k = c*9 + ky*3 + kx ; tile layout [c][row][col] with row stride 18, c stride 612.
// k >= 27 maps to a valid dummy slot (its A-column is all zeros, 0*finite = 0).
constexpr int koff(int k) {
  return (k >= 27) ? 0 : ((k / 9) * 34 + (k % 9) / 3) * 18 + (k % 9) % 3;
}

// ---------- radix-select pass 1: histogram of key[31:21] ----------
__global__ __launch_bounds__(256) void q_hist1(const float* __restrict__ x0,
                                               const float* __restrict__ x1,
                                               const float* __restrict__ x2,
                                               const float* __restrict__ x3,
                                               uint32_t* __restrict__ h1) {
  __shared__ uint32_t lh[2048];
  int tid = threadIdx.x;
  for (int i = tid; i < 2048; i += 256) lh[i] = 0;
  __syncthreads();
  int ch = blockIdx.x >> 2, part = blockIdx.x & 3;
  int img = ch / 48, bc = ch - img * 48;
  const float4* p = (const float4*)(pick(x0, x1, x2, x3, img) + (size_t)bc * HW) + part * 16384;
  for (int i = 0; i < 64; i++) {
    float4 v = p[i * 256 + tid];
    atomicAdd(&lh[fkey(v.x) >> 21], 1u);
    atomicAdd(&lh[fkey(v.y) >> 21], 1u);
    atomicAdd(&lh[fkey(v.z) >> 21], 1u);
    atomicAdd(&lh[fkey(v.w) >> 21], 1u);
  }
  __syncthreads();
  uint32_t* g = h1 + (size_t)ch * 2048;
  for (int i = tid; i < 2048; i += 256)
    if (lh[i]) atomicAdd(&g[i], lh[i]);
}

__global__ void q_scan1(const uint32_t* __restrict__ h1, uint32_t* __restrict__ selp,
                        uint32_t* __restrict__ selr) {
  int ch = blockIdx.x, r = threadIdx.x;
  if (r >= 4) return;
  uint32_t k = (r < 2) ? (2621u + r) : (23590u + r);  // 2621,2622,23592,23593
  const uint32_t* h = h1 + (size_t)ch * 2048;
  uint32_t cum = 0; int bin = 0;
  for (; bin < 2048; bin++) { uint32_t c = h[bin]; if (cum + c > k) break; cum += c; }
  selp[ch * 4 + r] = (uint32_t)bin;
  selr[ch * 4 + r] = k - cum;
}

// ---------- pass 2: conditional histogram of key[20:10] ----------
__global__ __launch_bounds__(256) void q_hist2(const float* __restrict__ x0,
                                               const float* __restrict__ x1,
                                               const float* __restrict__ x2,
                                               const float* __restrict__ x3,
                                               const uint32_t* __restrict__ selp,
                                               uint32_t* __restrict__ h2) {
  __shared__ uint32_t lh[8192];
  int tid = threadIdx.x;
  for (int i = tid; i < 8192; i += 256) lh[i] = 0;
  int ch = blockIdx.x >> 2, part = blockIdx.x & 3;
  int img = ch / 48, bc = ch - img * 48;
  uint32_t p0 = selp[ch * 4 + 0], p1 = selp[ch * 4 + 1];
  uint32_t p2 = selp[ch * 4 + 2], p3 = selp[ch * 4 + 3];
  __syncthreads();
  const float4* p = (const float4*)(pick(x0, x1, x2, x3, img) + (size_t)bc * HW) + part * 16384;
  for (int i = 0; i < 64; i++) {
    float4 v = p[i * 256 + tid];
#define DO2(val)                                                     \
    { uint32_t kk = fkey(val); uint32_t top = kk >> 21;              \
      uint32_t mid = (kk >> 10) & 2047u;                             \
      if (top == p0) atomicAdd(&lh[mid], 1u);                        \
      if (top == p1) atomicAdd(&lh[2048 + mid], 1u);                 \
      if (top == p2) atomicAdd(&lh[4096 + mid], 1u);                 \
      if (top == p3) atomicAdd(&lh[6144 + mid], 1u); }
    DO2(v.x) DO2(v.y) DO2(v.z) DO2(v.w)
#undef DO2
  }
  __syncthreads();
  uint32_t* g = h2 + (size_t)ch * 8192;
  for (int i = tid; i < 8192; i += 256)
    if (lh[i]) atomicAdd(&g[i], lh[i]);
}

__global__ void q_scan2(const uint32_t* __restrict__ h2, uint32_t* __restrict__ selp,
                        uint32_t* __restrict__ selr) {
  int ch = blockIdx.x, r = threadIdx.x;
  if (r >= 4) return;
  uint32_t k = selr[ch * 4 + r];
  const uint32_t* h = h2 + (size_t)ch * 8192 + r * 2048;
  uint32_t cum = 0; int bin = 0;
  for (; bin < 2048; bin++) { uint32_t c = h[bin]; if (cum + c > k) break; cum += c; }
  selp[ch * 4 + r] = (selp[ch * 4 + r] << 11) | (uint32_t)bin;  // 22-bit prefix
  selr[ch * 4 + r] = k - cum;
}

// ---------- pass 3: conditional histogram of key[9:0] ----------
__global__ __launch_bounds__(256) void q_hist3(const float* __restrict__ x0,
                                               const float* __restrict__ x1,
                                               const float* __restrict__ x2,
                                               const float* __restrict__ x3,
                                               const uint32_t* __restrict__ selp,
                                               uint32_t* __restrict__ h3) {
  __shared__ uint32_t lh[4096];
  int tid = threadIdx.x;
  for (int i = tid; i < 4096; i += 256) lh[i] = 0;
  int ch = blockIdx.x >> 2, part = blockIdx.x & 3;
  int img = ch / 48, bc = ch - img * 48;
  uint32_t p0 = selp[ch * 4 + 0], p1 = selp[ch * 4 + 1];
  uint32_t p2 = selp[ch * 4 + 2], p3 = selp[ch * 4 + 3];
  __syncthreads();
  const float4* p = (const float4*)(pick(x0, x1, x2, x3, img) + (size_t)bc * HW) + part * 16384;
  for (int i = 0; i < 64; i++) {
    float4 v = p[i * 256 + tid];
#define DO3(val)                                                    \
    { uint32_t kk = fkey(val); uint32_t hi22 = kk >> 10;            \
      uint32_t lo = kk & 1023u;                                     \
      if (hi22 == p0) atomicAdd(&lh[lo], 1u);                       \
      if (hi22 == p1) atomicAdd(&lh[1024 + lo], 1u);                \
      if (hi22 == p2) atomicAdd(&lh[2048 + lo], 1u);                \
      if (hi22 == p3) atomicAdd(&lh[3072 + lo], 1u); }
    DO3(v.x) DO3(v.y) DO3(v.z) DO3(v.w)
#undef DO3
  }
  __syncthreads();
  uint32_t* g = h3 + (size_t)ch * 4096;
  for (int i = tid; i < 4096; i += 256)
    if (lh[i]) atomicAdd(&g[i], lh[i]);
}

__global__ void q_scan3(const uint32_t* __restrict__ h3, const uint32_t* __restrict__ selp,
                        const uint32_t* __restrict__ selr, float* __restrict__ wbmul,
                        float* __restrict__ wbsub) {
  __shared__ float vq[4];
  int ch = blockIdx.x, r = threadIdx.x;
  if (r < 4) {
    uint32_t k = selr[ch * 4 + r];
    uint32_t pfx = selp[ch * 4 + r];
    const uint32_t* h = h3 + (size_t)ch * 4096 + r * 1024;
    uint32_t cum = 0; int bin = 0;
    for (; bin < 1024; bin++) { uint32_t c = h[bin]; if (cum + c > k) break; cum += c; }
    vq[r] = keyval((pfx << 10) | (uint32_t)bin);  // exact order statistic
  }
  __syncthreads();
  if (threadIdx.x == 0) {
    int img = ch / 48;
    float frac1 = 0.01f * 262143.0f - 2621.0f;
    float frac2 = 0.09f * 262143.0f - 23592.0f;
    float tmi = vq[0] + frac1 * (vq[1] - vq[0]);
    float tma = vq[2] + frac2 * (vq[3] - vq[2]);
    float d = tma - tmi;
    float pmi = (img < 2) ? 0.05f : 0.15f;
    float pma = (img == 1) ? 0.15f : ((img == 3) ? 0.20f : 0.10f);
    float tpmi = tmi - pmi * d;
    float tpma = tma + pma * d;
    wbmul[ch] = 1.0f / (tpma - tpmi + 1e-10f);
    wbsub[ch] = tpmi;
  }
}

// ---------- WMMA implicit-GEMM conv ----------
// White balance is folded into the GEMM: A' = w*mul_c, bias[o] = sum_k w*mul_c*sub_c,
// halo holds raw value sub_c so (sub_c - sub_c)*mul = 0 == SAME zero-padding of xw.
// Tile is then a pure byte copy of raw input -> async global->LDS DMA when available.
template <bool STATS>
__global__ __launch_bounds__(256) void conv_kernel(
    const float* __restrict__ x0, const float* __restrict__ x1,
    const float* __restrict__ x2, const float* __restrict__ x3,
    const float* __restrict__ cw, const float* __restrict__ wbmul,
    const float* __restrict__ wbsub, const float* __restrict__ gamma,
    const float* __restrict__ beta, const float* __restrict__ alpha,
    const float* __restrict__ meanArr, const float* __restrict__ rsArr,
    float* __restrict__ part1, float* __restrict__ part2, float* __restrict__ out) {
  __shared__ float tile[1836];  // 3 ch x 34 rows x 18 cols, RAW input (halo = sub_c)
  __shared__ float wlds[448];   // 16 x 28 scaled weights; rows o>=3 and col k=27 zero
  __shared__ float sbias[4];
  __shared__ float wpart[16];
  int tid = threadIdx.x;
  int bid = blockIdx.x;
  int txt = bid & 31, tyt = (bid >> 5) & 15, b = (bid >> 9) & 15, img = bid >> 13;
  int x0t = txt * 16, y0t = tyt * 32;
  const float* xp = pick(x0, x1, x2, x3, img);
  int chb = img * 48 + b * 3;
  float mul0 = wbmul[chb], mul1 = wbmul[chb + 1], mul2 = wbmul[chb + 2];
  float sub0 = wbsub[chb], sub1 = wbsub[chb + 1], sub2 = wbsub[chb + 2];
  // scaled weights: wlds[o][k] = w[o][k] * mul_{c(k)}
  for (int i = tid; i < 448; i += 256) {
    int o = i / 28, k = i - o * 28;
    float wv = 0.0f;
    if (o < 3 && k < 27) {
      int c = k / 9;
      float m = (c == 0) ? mul0 : ((c == 1) ? mul1 : mul2);
      wv = cw[(img * 3 + o) * 27 + k] * m;
    }
    wlds[i] = wv;
  }
  // raw tile copy (async DMA path when the gfx1250 builtin exists)
  for (int e = tid; e < 1836; e += 256) {
    int c = e / 612; int rr = e - c * 612;
    int tyy = rr / 18; int txx = rr - tyy * 18;
    int gy = y0t + tyy - 1, gx = x0t + txx - 1;
    bool inb = (gy >= 0 && gy < 512 && gx >= 0 && gx < 512);
    float s = (c == 0) ? sub0 : ((c == 1) ? sub1 : sub2);
#ifdef HAVE_ASYNC_LDS
    if (inb) {
      const float* gp = xp + (size_t)(b * 3 + c) * HW + gy * IMG_W + gx;
      __builtin_amdgcn_global_load_async_to_lds_b32(
          (AS1 int*)gp, (AS3 int*)&tile[e], 0, 0);
    } else {
      tile[e] = s;
    }
#else
    tile[e] = inb ? xp[(size_t)(b * 3 + c) * HW + gy * IMG_W + gx] : s;
#endif
  }
#ifdef HAVE_ASYNC_LDS
  __builtin_amdgcn_s_wait_asynccnt(0);
#endif
  __syncthreads();
  // bias[o] = sum_k wlds[o][k] * sub_{c(k)}
  if (tid < 3) {
    float bacc = 0.0f;
    for (int k = 0; k < 27; k++) {
      int c = k / 9;
      float s = (c == 0) ? sub0 : ((c == 1) ? sub1 : sub2);
      bacc += wlds[tid * 28 + k] * s;
    }
    sbias[tid] = bacc;
  }
  if (tid == 3) sbias[3] = 0.0f;
  __syncthreads();
  float bias0 = sbias[0], bias1 = sbias[1], bias2 = sbias[2];

  int lane = tid & 31, wv = tid >> 5;
  int o = lane & 15, hi = lane >> 4, px = lane & 15;
  v2f a[7];
#pragma unroll
  for (int t = 0; t < 7; t++) {
    int kb = o * 28 + 4 * t + 2 * hi;  // wlds fully zero-padded: unconditional loads
    a[t].x = wlds[kb];
    a[t].y = wlds[kb + 1];
  }
  int boff[14];
#pragma unroll
  for (int i2 = 0; i2 < 14; i2++) {
    const int t = i2 >> 1, j = i2 & 1;
    const int off_lo = koff(4 * t + j);      // constexpr
    const int off_hi = koff(4 * t + 2 + j);  // constexpr
    boff[i2] = px + (hi ? off_hi : off_lo);
  }

  float mean = 0.f, rs = 0.f, g0 = 0.f, g1 = 0.f, g2 = 0.f;
  float bt0 = 0.f, bt1 = 0.f, bt2 = 0.f, al0 = 0.f, al1 = 0.f, al2 = 0.f;
  if (!STATS) {
    mean = meanArr[img * 16 + b]; rs = rsArr[img * 16 + b];
    g0 = gamma[img * 3]; g1 = gamma[img * 3 + 1]; g2 = gamma[img * 3 + 2];
    bt0 = beta[img * 3]; bt1 = beta[img * 3 + 1]; bt2 = beta[img * 3 + 2];
    al0 = alpha[img * 3]; al1 = alpha[img * 3 + 1]; al2 = alpha[img * 3 + 2];
  }

  const float* tb = tile + wv * 72;  // this wave's 4-row band
  float lm = (lane < 16) ? 1.0f : 0.0f;  // lanes 16-31 carry zero rows of D
  float s1 = 0.0f, s2 = 0.0f;
#pragma unroll
  for (int r4 = 0; r4 < 4; r4++) {
    const int rb = r4 * 18;  // folds into ds_load immediate offset
    v8f acc = {0.f, 0.f, 0.f, 0.f, 0.f, 0.f, 0.f, 0.f};
#pragma unroll
    for (int t = 0; t < 7; t++) {
      v2f bf;
      bf.x = tb[boff[2 * t] + rb];
      bf.y = tb[boff[2 * t + 1] + rb];
      acc = __builtin_amdgcn_wmma_f32_16x16x4_f32(false, a[t], false, bf,
                                                  (short)0, acc, false, false);
    }
    if (STATS) {
      float f0 = (acc[0] - bias0) * lm;
      float f1 = (acc[1] - bias1) * lm;
      float f2 = (acc[2] - bias2) * lm;
      s1 += f0 + f1 + f2;
      s2 += f0 * f0 + f1 * f1 + f2 * f2;
    } else {
      if (lane < 16) {
        int gy = y0t + wv * 4 + r4, gx = x0t + px;
        float y0v = acc[0] - bias0, y1v = acc[1] - bias1, y2v = acc[2] - bias2;
        float t0 = (y0v - mean) * rs * g0 + bt0; t0 = (t0 >= 0.f) ? t0 : al0 * t0;
        float t1 = (y1v - mean) * rs * g1 + bt1; t1 = (t1 >= 0.f) ? t1 : al1 * t1;
        float t2 = (y2v - mean) * rs * g2 + bt2; t2 = (t2 >= 0.f) ? t2 : al2 * t2;
        size_t ob = (((size_t)b * 12 + img * 3) * 512 + gy) * 512 + gx;
        out[ob] = t0;
        out[ob + (size_t)HW] = t1;
        out[ob + 2 * (size_t)HW] = t2;
      }
    }
  }
  if (STATS) {
#pragma unroll
    for (int off = 16; off > 0; off >>= 1) {
      s1 += __shfl_down(s1, off);
      s2 += __shfl_down(s2, off);
    }
    if (lane == 0) { wpart[wv * 2] = s1; wpart[wv * 2 + 1] = s2; }
    __syncthreads();
    if (tid == 0) {
      float t1 = 0.f, t2 = 0.f;
      for (int w = 0; w < 8; w++) { t1 += wpart[w * 2]; t2 += wpart[w * 2 + 1]; }
      part1[bid] = t1; part2[bid] = t2;  // deterministic per-block partial
    }
  }
}

// ---------- deterministic reduction of 512 tile partials per (img,b) ----------
__global__ __launch_bounds__(256) void stats_reduce(const float* __restrict__ part1,
                                                    const float* __restrict__ part2,
                                                    float* __restrict__ meanArr,
                                                    float* __restrict__ rsArr) {
  __shared__ float sh1[256], sh2[256];
  int ib = blockIdx.x;        // img*16 + b
  int img = ib >> 4, b = ib & 15;
  int base = (img << 13) | (b << 9);
  int tid = threadIdx.x;
  sh1[tid] = part1[base + tid] + part1[base + 256 + tid];
  sh2[tid] = part2[base + tid] + part2[base + 256 + tid];
  __syncthreads();
  for (int off = 128; off > 0; off >>= 1) {
    if (tid < off) { sh1[tid] += sh1[tid + off]; sh2[tid] += sh2[tid + off]; }
    __syncthreads();
  }
  if (tid == 0) {
    float n = 786432.0f;  // 3*512*512
    float mean = sh1[0] / n;
    float var = sh2[0] / n - mean * mean;
    meanArr[ib] = mean;
    rsArr[ib] = rsqrtf(var + 1e-5f);
  }
}

extern "C" void kernel_launch(void* const* d_in, const int* in_sizes, int n_in,
                              void* d_out, int out_size, void* d_ws, size_t ws_size,
                              hipStream_t stream) {
  const float* x0 = (const float*)d_in[0];
  const float* x1 = (const float*)d_in[1];
  const float* x2 = (const float*)d_in[2];
  const float* x3 = (const float*)d_in[3];
  const float* cw = (const float*)d_in[4];
  const float* gamma = (const float*)d_in[5];
  const float* beta = (const float*)d_in[6];
  const float* alpha = (const float*)d_in[7];
  float* out = (float*)d_out;
  char* ws = (char*)d_ws;
  uint32_t* h1 = (uint32_t*)(ws + OFF_H1);
  uint32_t* h2 = (uint32_t*)(ws + OFF_H2);
  uint32_t* h3 = (uint32_t*)(ws + OFF_H3);
  uint32_t* selp = (uint32_t*)(ws + OFF_SELP);
  uint32_t* selr = (uint32_t*)(ws + OFF_SELR);
  float* wbm = (float*)(ws + OFF_WBM);
  float* wbs = (float*)(ws + OFF_WBS);
  float* meanA = (float*)(ws + OFF_MEAN);
  float* rsA = (float*)(ws + OFF_RS);
  float* p1 = (float*)(ws + OFF_P1);
  float* p2 = (float*)(ws + OFF_P2);

  (void)hipMemsetAsync(ws, 0, OFF_ZEND, stream);  // zero all histograms every call

  q_hist1<<<768, 256, 0, stream>>>(x0, x1, x2, x3, h1);
  q_scan1<<<192, 32, 0, stream>>>(h1, selp, selr);
  q_hist2<<<768, 256, 0, stream>>>(x0, x1, x2, x3, selp, h2);
  q_scan2<<<192, 32, 0, stream>>>(h2, selp, selr);
  q_hist3<<<768, 256, 0, stream>>>(x0, x1, x2, x3, selp, h3);
  q_scan3<<<192, 32, 0, stream>>>(h3, selp, selr, wbm, wbs);

  conv_kernel<true><<<32768, 256, 0, stream>>>(x0, x1, x2, x3, cw, wbm, wbs, gamma,
                                               beta, alpha, meanA, rsA, p1, p2, out);
  stats_reduce<<<64, 256, 0, stream>>>(p1, p2, meanA, rsA);
  conv_kernel<false><<<32768, 256, 0, stream>>>(x0, x1, x2, x3, cw, wbm, wbs, gamma,
                                                beta, alpha, meanA, rsA, p1, p2, out);
}